// QueryAndGroup_47665547051518
// MI455X (gfx1250) — compile-verified
//
#include <hip/hip_runtime.h>
#include <hip/hip_bf16.h>

#define N_POINTS 65536
#define NQ       4096
#define FEAT     67
#define NSAMPLE  16
#define WAVES_PER_BLOCK 8

typedef float v2f __attribute__((ext_vector_type(2)));
typedef float v4f __attribute__((ext_vector_type(4)));
typedef float v8f __attribute__((ext_vector_type(8)));

// ---------------------------------------------------------------------------
// Kernel 1: build augmented point array: aug[j] = {x, y, z, x^2+y^2+z^2}
// ---------------------------------------------------------------------------
__global__ void qg_prep_kernel(const float* __restrict__ feats,
                               float* __restrict__ aug) {
    int j = blockIdx.x * blockDim.x + threadIdx.x;
    if (j >= N_POINTS) return;
    float x = feats[j * FEAT + 0];
    float y = feats[j * FEAT + 1];
    float z = feats[j * FEAT + 2];
    v4f o = {x, y, z, x * x + y * y + z * z};
    *(v4f*)(&aug[j * 4]) = o;
}

// Register-resident top-16 (smallest) insertion. All indices compile-time so
// arrays live in VGPRs (conditional writes lower to v_cndmask).
__device__ __forceinline__ void insert16(float d, int i,
                                         float (&bd)[16], int (&bi)[16],
                                         float& worst, int& wpos) {
    if (d < worst) {
#pragma unroll
        for (int t = 0; t < 16; ++t)
            if (t == wpos) { bd[t] = d; bi[t] = i; }
        float w = bd[0]; int wp = 0;
#pragma unroll
        for (int t = 1; t < 16; ++t)
            if (bd[t] > w) { w = bd[t]; wp = t; }
        worst = w; wpos = wp;
    }
}

// ---------------------------------------------------------------------------
// Kernel 2: per block: 16 queries. 8 waves, each wave scans 8192 points in
// 2x16-point tiles per iteration via V_WMMA_F32_16X16X4_F32
// (relative d2 = -2 q.p + |p|^2), software-pipelined B loads, transpose of
// the D tiles through wave-private LDS, per-lane register top-16, final
// per-query merge of 256 candidates through LDS.
// ---------------------------------------------------------------------------
__global__ __launch_bounds__(256) void qg_knn_wmma_kernel(
        const float* __restrict__ aug, int* __restrict__ idx_sel) {
    __shared__ float tileLds[WAVES_PER_BLOCK * 512]; // 2x(16x16 f32) per wave
    __shared__ float candD[16 * 256];
    __shared__ int   candI[16 * 256];

    const int lane = threadIdx.x & 31;
    const int wave = threadIdx.x >> 5;
    const int h    = lane >> 4;   // half select (K-pair for A/B, row-half for D)
    const int l15  = lane & 15;
    const int qBase = blockIdx.x * 16;

    // A operand: row m = query qBase+l15, values [-2qx,-2qy,-2qz,1].
    const int q = qBase + l15;
    v2f a;
    a.x = -2.0f * aug[q * 4 + 2 * h + 0];
    a.y = h ? 1.0f : (-2.0f * aug[q * 4 + 1]);

    float bd[16]; int bi[16];
#pragma unroll
    for (int t = 0; t < 16; ++t) { bd[t] = 3.0e38f; bi[t] = 0; }
    float worst = 3.0e38f; int wpos = 0;

    float* tw0 = &tileLds[wave * 512];
    float* tw1 = tw0 + 256;
    const int ptsPerWave = N_POINTS / WAVES_PER_BLOCK; // 8192
    const int base0 = wave * ptsPerWave;
    const int nIter = ptsPerWave / 32;                 // 256

    // Software pipeline: preload first pair of B operands.
    const int boff = l15 * 4 + 2 * h;
    v2f b0 = *(const v2f*)(&aug[base0 * 4 + boff]);
    v2f b1 = *(const v2f*)(&aug[(base0 + 16) * 4 + boff]);

    int tb = base0;
    for (int it = 0; it < nIter; ++it, tb += 32) {
        // Prefetch well ahead of the pipelined loads (L2-resident stream).
        __builtin_prefetch(&aug[(tb + 160) * 4], 0, 3);

        // Preload next iteration's B operands (wave-uniform clamp at the end
        // so we never read past the wave's range / end of aug).
        const int nbase = (it + 1 < nIter) ? (tb + 32) : base0;
        v2f nb0 = *(const v2f*)(&aug[nbase * 4 + boff]);
        v2f nb1 = *(const v2f*)(&aug[(nbase + 16) * 4 + boff]);

        v8f c0 = {0.f, 0.f, 0.f, 0.f, 0.f, 0.f, 0.f, 0.f};
        v8f d0 = __builtin_amdgcn_wmma_f32_16x16x4_f32(
            false, a, false, b0, (short)0, c0, false, false);
        v8f c1 = {0.f, 0.f, 0.f, 0.f, 0.f, 0.f, 0.f, 0.f};
        v8f d1 = __builtin_amdgcn_wmma_f32_16x16x4_f32(
            false, a, false, b1, (short)0, c1, false, false);

        // Scatter both D tiles: lane holds rows 8h..8h+7 of column l15.
#pragma unroll
        for (int r = 0; r < 8; ++r)
            tw0[(8 * h + r) * 16 + l15] = d0[r];
#pragma unroll
        for (int r = 0; r < 8; ++r)
            tw1[(8 * h + r) * 16 + l15] = d1[r];

        // Re-read transposed: this lane owns query row l15, columns 8h..8h+7
        // of both tiles. Same-wave LDS ops are in-order (ISA ch11).
        const float* rp0 = &tw0[l15 * 16 + 8 * h];
        const float* rp1 = &tw1[l15 * 16 + 8 * h];
        v4f g0 = *(const v4f*)(rp0);
        v4f g1 = *(const v4f*)(rp0 + 4);
        v4f g2 = *(const v4f*)(rp1);
        v4f g3 = *(const v4f*)(rp1 + 4);
#pragma unroll
        for (int t = 0; t < 4; ++t)
            insert16(g0[t], tb + 8 * h + t, bd, bi, worst, wpos);
#pragma unroll
        for (int t = 0; t < 4; ++t)
            insert16(g1[t], tb + 8 * h + 4 + t, bd, bi, worst, wpos);
#pragma unroll
        for (int t = 0; t < 4; ++t)
            insert16(g2[t], tb + 16 + 8 * h + t, bd, bi, worst, wpos);
#pragma unroll
        for (int t = 0; t < 4; ++t)
            insert16(g3[t], tb + 16 + 8 * h + 4 + t, bd, bi, worst, wpos);

        b0 = nb0;
        b1 = nb1;
    }

    // Dump per-lane lists: query row l15 gets slot block (wave*2+h)*16.
    {
        const int slot = (wave * 2 + h) * 16;
        float* cd = &candD[l15 * 256 + slot];
        int*   ci = &candI[l15 * 256 + slot];
#pragma unroll
        for (int t = 0; t < 16; ++t) { cd[t] = bd[t]; ci[t] = bi[t]; }
    }
    __syncthreads();

    // Final merge: one thread per query scans 256 candidates, then emits the
    // 16 winners in ascending-d2 order (matches top_k(-d2) ordering).
    if (threadIdx.x < 16) {
        const int qr = threadIdx.x;
        float fd[16]; int fi[16];
#pragma unroll
        for (int t = 0; t < 16; ++t) { fd[t] = 3.0e38f; fi[t] = 0; }
        float w2 = 3.0e38f; int wp2 = 0;
        const float* cd = &candD[qr * 256];
        const int*   ci = &candI[qr * 256];
        for (int j = 0; j < 256; ++j)
            insert16(cd[j], ci[j], fd, fi, w2, wp2);

        bool used[16];
#pragma unroll
        for (int t = 0; t < 16; ++t) used[t] = false;
        int* outIdx = &idx_sel[(qBase + qr) * NSAMPLE];
#pragma unroll
        for (int k = 0; k < 16; ++k) {
            float best = 3.4e38f; int bpos = 0;
#pragma unroll
            for (int t = 0; t < 16; ++t)
                if (!used[t] && fd[t] < best) { best = fd[t]; bpos = t; }
            int sel = 0;
#pragma unroll
            for (int t = 0; t < 16; ++t)
                if (t == bpos) { used[t] = true; sel = fi[t]; }
            outIdx[k] = sel;
        }
    }
}

// ---------------------------------------------------------------------------
// Kernel 3: gather. out[(q*16+k)*67 + c] = feats[idx,c] - (c<3 ? feats[q,c]:0)
// ---------------------------------------------------------------------------
__global__ void qg_gather_kernel(const float* __restrict__ feats,
                                 const int* __restrict__ idx_sel,
                                 float* __restrict__ out) {
    int gid = blockIdx.x * blockDim.x + threadIdx.x;
    const int total = NQ * NSAMPLE * FEAT;
    if (gid >= total) return;
    int c = gid % FEAT;
    int e = gid / FEAT;   // q*16 + k
    int q = e >> 4;
    int idx = idx_sel[e];
    float v = feats[idx * FEAT + c];
    if (c < 3) v -= feats[q * FEAT + c];
    out[gid] = v;
}

extern "C" void kernel_launch(void* const* d_in, const int* in_sizes, int n_in,
                              void* d_out, int out_size, void* d_ws, size_t ws_size,
                              hipStream_t stream) {
    const float* feats = (const float*)d_in[0];
    float* out = (float*)d_out;

    // Workspace layout: [ aug: 65536*4 f32 = 1 MiB ][ idx_sel: 4096*16 i32 ]
    float* aug = (float*)d_ws;
    int* idx_sel = (int*)((char*)d_ws + (size_t)N_POINTS * 4 * sizeof(float));

    qg_prep_kernel<<<N_POINTS / 256, 256, 0, stream>>>(feats, aug);
    qg_knn_wmma_kernel<<<NQ / 16, 256, 0, stream>>>(aug, idx_sel);

    const int total = NQ * NSAMPLE * FEAT;
    qg_gather_kernel<<<(total + 255) / 256, 256, 0, stream>>>(feats, idx_sel, out);
}